// GranularityAwarenessSensor_55594056679884
// MI455X (gfx1250) — compile-verified
//
#include <hip/hip_runtime.h>
#include <hip/hip_bf16.h>

// Problem constants (from reference setup)
#define BB   16
#define HH   12
#define NN   576
#define NP1  577
#define DD   768
#define KSEL 45
#define WPAD (2 * KSEL)   // 90 gathered rows

typedef __attribute__((ext_vector_type(2))) float v2f;
typedef __attribute__((ext_vector_type(8))) float v8f;

// ---------------------------------------------------------------------------
// 1) A_cls[b,n] = mean_h attention_scores[b,h,0,1+n]
// ---------------------------------------------------------------------------
__global__ void acls_kernel(const float* __restrict__ att, float* __restrict__ acls) {
    int idx = blockIdx.x * 256 + threadIdx.x;
    if (idx >= BB * NN) return;
    int b = idx / NN, n = idx % NN;
    const size_t head_sz = (size_t)NP1 * NP1;
    const float* base = att + (size_t)b * HH * head_sz + (size_t)(n + 1);
    float s = 0.0f;
#pragma unroll
    for (int h = 0; h < HH; ++h) s += base[(size_t)h * head_sz];
    acls[idx] = s / (float)HH;
}

// ---------------------------------------------------------------------------
// 2) Per-batch top-45 (iterative argmax, lowest-index tie-break, matching
//    jax.lax.top_k ordering) + initialize mask0.
// ---------------------------------------------------------------------------
__global__ void topk_mask_kernel(const float* __restrict__ acls,
                                 int* __restrict__ topk, float* __restrict__ mask) {
    __shared__ float vals[NN];
    __shared__ float rv[256];
    __shared__ int   ri[256];
    int b = blockIdx.x, t = threadIdx.x;
    for (int n = t; n < NN; n += 256) {
        vals[n] = acls[b * NN + n];
        mask[b * NN + n] = 0.0f;
    }
    __syncthreads();
    for (int i = 0; i < KSEL; ++i) {
        float bv = -__builtin_inff();
        int   bi = 0x7fffffff;
        for (int n = t; n < NN; n += 256) {
            float v = vals[n];
            bool better = (v > bv) | ((v == bv) & (n < bi));
            bv = better ? v : bv;
            bi = better ? n : bi;
        }
        rv[t] = bv; ri[t] = bi;
        __syncthreads();
        for (int s = 128; s > 0; s >>= 1) {
            if (t < s) {
                float ov = rv[t + s]; int oi = ri[t + s];
                bool better = (ov > rv[t]) | ((ov == rv[t]) & (oi < ri[t]));
                rv[t] = better ? ov : rv[t];
                ri[t] = better ? oi : ri[t];
            }
            __syncthreads();
        }
        if (t == 0) {
            int sel = ri[0];
            topk[b * KSEL + i] = sel;
            mask[b * NN + sel] = 1.0f;
            vals[sel] = -__builtin_inff();
        }
        __syncthreads();
    }
}

// ---------------------------------------------------------------------------
// 3a) One greedy step: per-batch masked-similarity argmax via
//     V_WMMA_F32_16X16X4_F32 (exact f32 matrix pipe).
//     Block b handles batch b: 36 row-tiles of 16 patches, K looped 4-wide.
// ---------------------------------------------------------------------------
__global__ void sim_argmax_kernel(const float* __restrict__ F,
                                  const float* __restrict__ mask,
                                  const int* __restrict__ topk, int step,
                                  float* __restrict__ bestv_out,
                                  int* __restrict__ besti_out) {
    __shared__ float qs[DD];
    __shared__ float wbv[8];
    __shared__ int   wbi[8];

    const int b = blockIdx.x;
    const int t = threadIdx.x;
    const int lane = t & 31;
    const int wave = t >> 5;
    const int m  = lane & 15;   // A-matrix row within tile (M)
    const int kh = lane >> 4;   // 0: K={0,1}  1: K={2,3}

    const float* Fb = F + (size_t)b * NN * DD;
    const int qi = topk[b * KSEL + step];
    const float* qrow = Fb + (size_t)qi * DD;
    for (int d = t; d < DD; d += 256) qs[d] = qrow[d];
    __syncthreads();

    const float* maskb = mask + b * NN;
    float bestv = -__builtin_inff();
    int   besti = 0x7fffffff;

    for (int tile = wave; tile < NN / 16; tile += 8) {   // wave-uniform bound
        const int n0 = tile * 16;
        const float* arow = Fb + (size_t)(n0 + m) * DD + 2 * kh;
        v8f acc = {};
#pragma unroll 8
        for (int k = 0; k < DD; k += 4) {
            v2f a  = *(const v2f*)(arow + k);          // A: F[n0+m, k+2*kh .. +1]
            v2f bq = *(const v2f*)(&qs[k + 2 * kh]);   // B: q replicated over N cols
            acc = __builtin_amdgcn_wmma_f32_16x16x4_f32(
                false, a, false, bq, (short)0, acc, false, false);
        }
        // C layout: VGPR j holds row (n0 + j + 8*kh); all 16 columns identical.
        // Prefetch the 8 mask values (independent loads -> one clause/wait),
        // then branchless argmax (v_cndmask, no EXEC gymnastics).
        float mv[8];
#pragma unroll
        for (int j = 0; j < 8; ++j) mv[j] = maskb[n0 + kh * 8 + j];
#pragma unroll
        for (int j = 0; j < 8; ++j) {
            int n = n0 + kh * 8 + j;
            float s = acc[j] * (1.0f - mv[j]);          // sim * (1 - mask)
            bool better = (s > bestv) | ((s == bestv) & (n < besti));
            bestv = better ? s : bestv;
            besti = better ? n : besti;
        }
    }

    // wave32 argmax reduction (tie-break: lowest index, = jnp.argmax)
    for (int off = 16; off > 0; off >>= 1) {
        float ov = __shfl_xor(bestv, off, 32);
        int   oi = __shfl_xor(besti, off, 32);
        bool better = (ov > bestv) | ((ov == bestv) & (oi < besti));
        bestv = better ? ov : bestv;
        besti = better ? oi : besti;
    }
    if (lane == 0) { wbv[wave] = bestv; wbi[wave] = besti; }
    __syncthreads();
    if (t == 0) {
        float v = wbv[0]; int i0 = wbi[0];
        for (int w = 1; w < 8; ++w) {
            bool better = (wbv[w] > v) | ((wbv[w] == v) & (wbi[w] < i0));
            v  = better ? wbv[w] : v;
            i0 = better ? wbi[w] : i0;
        }
        bestv_out[b] = v;
        besti_out[b] = i0;
    }
}

// ---------------------------------------------------------------------------
// 3b) Global scalar test (sim.max() over the WHOLE batch) + conditional
//     update of every batch's mask (reference semantics).
// ---------------------------------------------------------------------------
__global__ void update_kernel(const float* __restrict__ bv,
                              const int* __restrict__ bi,
                              float* __restrict__ mask) {
    if (threadIdx.x == 0) {
        float g = -__builtin_inff();
        for (int b = 0; b < BB; ++b) g = fmaxf(g, bv[b]);
        if (g > 0.0f) {
            for (int b = 0; b < BB; ++b) mask[b * NN + bi[b]] = 1.0f;
        }
    }
}

// ---------------------------------------------------------------------------
// 4) Stable gather: selected rows (ascending index, mask==1 → masked==F),
//    zero-padded to 90 rows.
// ---------------------------------------------------------------------------
__global__ void gather_kernel(const float* __restrict__ F,
                              const float* __restrict__ mask,
                              float* __restrict__ gath) {
    __shared__ int sel[WPAD];
    __shared__ int cnt_s;
    const int b = blockIdx.x, t = threadIdx.x;
    if (t == 0) {
        int c = 0;
        for (int n = 0; n < NN; ++n) {
            if (mask[b * NN + n] != 0.0f) { if (c < WPAD) sel[c] = n; ++c; }
        }
        cnt_s = c < WPAD ? c : WPAD;
    }
    __syncthreads();
    const int cnt = cnt_s;
    const int C = DD / 4;  // 192 float4 per row
    const float4* Fb = (const float4*)(F + (size_t)b * NN * DD);
    float4* Gb = (float4*)(gath + (size_t)b * WPAD * DD);
    for (int e = t; e < WPAD * C; e += 256) {
        int i = e / C, c = e % C;
        float4 v = make_float4(0.0f, 0.0f, 0.0f, 0.0f);
        if (i < cnt) v = Fb[(size_t)sel[i] * C + c];
        Gb[(size_t)i * C + c] = v;
    }
}

// ---------------------------------------------------------------------------
extern "C" void kernel_launch(void* const* d_in, const int* in_sizes, int n_in,
                              void* d_out, int out_size, void* d_ws, size_t ws_size,
                              hipStream_t stream) {
    const float* att = (const float*)d_in[0];  // [16,12,577,577] f32
    const float* F   = (const float*)d_in[1];  // [16,576,768]   f32

    float* mask = (float*)d_out;               // [16,576]       (output 0)
    float* gath = (float*)d_out + BB * NN;     // [16,90,768]    (output 1)

    // workspace carve-out (~40 KB)
    float* acls = (float*)d_ws;                            // 16*576 f32
    int*   topk = (int*)(acls + BB * NN);                  // 16*45  i32
    float* bv   = (float*)(topk + BB * KSEL);              // 16     f32
    int*   bi   = (int*)(bv + BB);                         // 16     i32

    acls_kernel<<<(BB * NN + 255) / 256, 256, 0, stream>>>(att, acls);
    topk_mask_kernel<<<BB, 256, 0, stream>>>(acls, topk, mask);
    for (int step = 0; step < KSEL; ++step) {
        sim_argmax_kernel<<<BB, 256, 0, stream>>>(F, mask, topk, step, bv, bi);
        update_kernel<<<1, 32, 0, stream>>>(bv, bi, mask);
    }
    gather_kernel<<<BB, 256, 0, stream>>>(F, mask, gath);
}